// GraphConv_86114094285068
// MI455X (gfx1250) — compile-verified
//
#include <hip/hip_runtime.h>

#define NN 10000
#define NE 640000
#define D  128

typedef __attribute__((ext_vector_type(2))) float v2f;
typedef __attribute__((ext_vector_type(8))) float v8f;

// ---------------- Phase 0: zero the aggregation buffer (in d_ws) ------------
__global__ void gc_zero_agg(float4* __restrict__ agg4, int n4) {
    int i = blockIdx.x * blockDim.x + threadIdx.x;
    if (i < n4) agg4[i] = make_float4(0.f, 0.f, 0.f, 0.f);
}

// ---------------- Phase 1: edge scatter with native f32 L2 atomics ----------
// One wave (32 lanes) per edge: lane l handles channels [4l, 4l+4).
// src/dst/val are wave-uniform -> scalar loads; x row read as float4 (L2 hit).
__global__ void gc_edge_scatter(const float* __restrict__ x,
                                const int*   __restrict__ src,
                                const int*   __restrict__ dst,
                                const float* __restrict__ vals,
                                float*       __restrict__ agg) {
    int e    = (blockIdx.x * blockDim.x + threadIdx.x) >> 5;
    int lane = threadIdx.x & 31;
    if (e >= NE) return;
    int   s = src[e];
    int   d = dst[e];
    float v = vals[e];
    float4 m = ((const float4*)(x + (size_t)s * D))[lane];
    float* a = agg + (size_t)d * D + lane * 4;
    unsafeAtomicAdd(a + 0, v * m.x);   // global_atomic_add_f32, no return
    unsafeAtomicAdd(a + 1, v * m.y);
    unsafeAtomicAdd(a + 2, v * m.z);
    unsafeAtomicAdd(a + 3, v * m.w);
}

// ---------------- Phase 2: out = agg @ W^T + b via V_WMMA_F32_16X16X4_F32 ---
// One wave per 16x16 output tile. 625 M-tiles x 8 N-tiles = 5000 waves =
// 625 blocks x 8 waves. K=128 in 32 steps of 4. All-lanes-active (EXEC==~0).
__global__ void gc_gemm_wmma(const float* __restrict__ agg,
                             const float* __restrict__ W,
                             const float* __restrict__ bias,
                             float*       __restrict__ out) {
    int wave  = blockIdx.x * 8 + (threadIdx.x >> 5);
    int lane  = threadIdx.x & 31;
    int mBase = (wave >> 3) * 16;      // node-tile
    int nBase = (wave & 7) * 16;       // output-channel tile
    int r15   = lane & 15;
    int hi    = lane >> 4;             // 0: K pair {0,1}, 1: K pair {2,3}

    // A frag: agg[mBase + (lane&15)][k + 2*hi .. +1]   (16x4, f32)
    // B frag: B[k][n] = W[n][k] -> W[nBase + (lane&15)][k + 2*hi .. +1]
    const float* aPtr = agg + (size_t)(mBase + r15) * D + 2 * hi;
    const float* bPtr = W   + (size_t)(nBase + r15) * D + 2 * hi;

    v8f c = {};
#pragma unroll
    for (int k = 0; k < D; k += 4) {
        v2f af = *(const v2f*)(aPtr + k);
        v2f bf = *(const v2f*)(bPtr + k);
        // (neg_a, A, neg_b, B, c_mod, C, reuse_a, reuse_b)
        c = __builtin_amdgcn_wmma_f32_16x16x4_f32(
                false, af, false, bf, (short)0, c, false, false);
    }

    // C/D layout: VGPR r -> M = r + 8*hi, N = lane&15
    float bb = bias[nBase + r15];
    float* o = out + (size_t)(mBase + 8 * hi) * D + nBase + r15;
#pragma unroll
    for (int r = 0; r < 8; ++r)
        o[(size_t)r * D] = c[r] + bb;
}

extern "C" void kernel_launch(void* const* d_in, const int* in_sizes, int n_in,
                              void* d_out, int out_size, void* d_ws, size_t ws_size,
                              hipStream_t stream) {
    (void)in_sizes; (void)n_in; (void)out_size; (void)ws_size;
    const float* x    = (const float*)d_in[0];
    const int*   src  = (const int*)  d_in[1];
    const int*   dst  = (const int*)  d_in[2];
    const float* vals = (const float*)d_in[3];
    const float* W    = (const float*)d_in[4];
    const float* b    = (const float*)d_in[5];
    float* out = (float*)d_out;
    float* agg = (float*)d_ws;                    // NN*D*4 = 5.12 MB scratch

    int n4 = NN * D / 4;                          // 320000 float4s
    gc_zero_agg<<<(n4 + 255) / 256, 256, 0, stream>>>((float4*)agg, n4);

    gc_edge_scatter<<<(NE + 7) / 8, 256, 0, stream>>>(x, src, dst, vals, agg);

    gc_gemm_wmma<<<(NN / 16) * 8 / 8, 256, 0, stream>>>(agg, W, b, out);
}